// DDI_76751065579531
// MI455X (gfx1250) — compile-verified
//
#include <hip/hip_runtime.h>
#include <hip/hip_bf16.h>

#define BB   256
#define NN   200
#define EE   400
#define DIN  360
#define DOUT 360
#define MROWS (BB * NN)            // 51200
#define ROWS_PER_BLOCK 128
#define CT_COUNT 12                // ceil(360/32)
#define KSTEPS (DIN / 4)           // 90

typedef float v2f __attribute__((ext_vector_type(2)));
typedef float v8f __attribute__((ext_vector_type(8)));

// ---------------------------------------------------------------------------
// Kernel 0: zero workspace (grid-stride, float4)
// ---------------------------------------------------------------------------
__global__ __launch_bounds__(256) void zero_ws_kernel(float4* __restrict__ p, long n4) {
    long i = (long)blockIdx.x * blockDim.x + threadIdx.x;
    long stride = (long)gridDim.x * blockDim.x;
    float4 z = {0.f, 0.f, 0.f, 0.f};
    for (; i < n4; i += stride) p[i] = z;
}

// ---------------------------------------------------------------------------
// Kernel 1: edge aggregation (linear part commuted before the GEMM).
//   Xagg_in [b, dst] += data * X[b, src]
//   Xagg_out[b, src] += data * X[b, dst]
//   deg_in[b, dst]  += data ;  deg_out[b, src] += data
// One wave32 per edge, fp32 HW atomics.
// ---------------------------------------------------------------------------
__global__ __launch_bounds__(256) void sp_agg_kernel(
    const float* __restrict__ X,
    const long long* __restrict__ adj,   // [B, E, 2] int64
    const float* __restrict__ adjd,      // [B, E]
    float* __restrict__ xagg_in,
    float* __restrict__ xagg_out,
    float* __restrict__ deg_in,
    float* __restrict__ deg_out)
{
    int gwave = (int)((blockIdx.x * blockDim.x + threadIdx.x) >> 5);  // wave32
    int lane  = threadIdx.x & 31;
    if (gwave >= BB * EE) return;
    int b = gwave / EE;
    int e = gwave - b * EE;

    long long src = adj[((long)b * EE + e) * 2 + 0];
    long long dst = adj[((long)b * EE + e) * 2 + 1];
    float dat = adjd[(long)b * EE + e];

    const float* xs = X + ((long)b * NN + src) * DIN;
    const float* xd = X + ((long)b * NN + dst) * DIN;
    float* ain  = xagg_in  + ((long)b * NN + dst) * DIN;
    float* aout = xagg_out + ((long)b * NN + src) * DIN;

    for (int d = lane; d < DIN; d += 32) {
        unsafeAtomicAdd(&ain[d],  dat * xs[d]);
        unsafeAtomicAdd(&aout[d], dat * xd[d]);
    }
    if (lane == 0) {
        unsafeAtomicAdd(&deg_in [(long)b * NN + dst], dat);
        unsafeAtomicAdd(&deg_out[(long)b * NN + src], dat);
    }
}

// ---------------------------------------------------------------------------
// Kernel 2: fused triple-GEMM + gates + epilogue, fp32 WMMA 16x16x4.
// Block = 256 threads (8 wave32) owns 128 rows. Source loop outermost:
// each wave loads its 16x360 fp32 A-tile ONCE into 180 VGPRs (wave32 VGPR
// file = 1024), then sweeps 12 32-wide column tiles with the weight chunk
// staged transposed in LDS. Inner step: 2x ds_load_b64 + 2x v_wmma — zero
// global A traffic inside the sweep (12x reduction in L2 traffic vs
// reloading A per tile). Three sources accumulate into `out` via per-lane
// RMW (L2-resident tile), final pass applies ReLU.
// out[row,col] = relu( sum_s (C_s + deg_s*b_s[col]) * sigmoid(gate_s[row]) )
// ---------------------------------------------------------------------------
__global__ __launch_bounds__(256, 1) void fused_gemm_kernel(
    const float* __restrict__ X,         // [M, DIN]  (loop source)
    const float* __restrict__ xagg_in,   // [M, DIN]
    const float* __restrict__ xagg_out,  // [M, DIN]
    const float* __restrict__ deg_in,    // [M]
    const float* __restrict__ deg_out,   // [M]
    const float* __restrict__ w_in,      // [DIN, DOUT]
    const float* __restrict__ b_in,      // [DOUT]
    const float* __restrict__ w_out,
    const float* __restrict__ b_out,
    const float* __restrict__ w_loop,
    const float* __restrict__ w_gin,     // [DIN]
    const float* __restrict__ b_gin,     // [1]
    const float* __restrict__ w_gout,
    const float* __restrict__ b_gout,
    const float* __restrict__ w_gloop,
    float* __restrict__ out)             // [M, DOUT]
{
    __shared__ float wlds[32 * DIN];           // transposed W chunk: 46,080 B
    __shared__ float s_sig[3][ROWS_PER_BLOCK]; // sigmoid(gate) per source
    __shared__ float s_deg[2][ROWS_PER_BLOCK]; // deg_in, deg_out

    const int tid  = threadIdx.x;
    const int wave = tid >> 5;
    const int lane = tid & 31;
    const int row0 = blockIdx.x * ROWS_PER_BLOCK;

    // ---- gate phase: wave-cooperative, coalesced ------------------------
    for (int r = 0; r < 16; ++r) {
        const int lr = wave * 16 + r;            // local row 0..127
        const long row = row0 + lr;
        const float* xi = xagg_in  + row * DIN;
        const float* xo = xagg_out + row * DIN;
        const float* xl = X        + row * DIN;
        float gin = 0.f, gout = 0.f, gloop = 0.f;
        for (int k = lane; k < DIN; k += 32) {
            gin   += xi[k] * w_gin[k];
            gout  += xo[k] * w_gout[k];
            gloop += xl[k] * w_gloop[k];
        }
        #pragma unroll
        for (int off = 16; off > 0; off >>= 1) {
            gin   += __shfl_xor(gin,   off, 32);
            gout  += __shfl_xor(gout,  off, 32);
            gloop += __shfl_xor(gloop, off, 32);
        }
        if (lane == 0) {
            float di = deg_in[row], dm = deg_out[row];
            s_sig[0][lr] = 1.f / (1.f + __expf(-(gin  + di * b_gin[0])));
            s_sig[1][lr] = 1.f / (1.f + __expf(-(gout + dm * b_gout[0])));
            s_sig[2][lr] = 1.f / (1.f + __expf(-gloop));
            s_deg[0][lr] = di;
            s_deg[1][lr] = dm;
        }
    }
    __syncthreads();

    // ---- WMMA phase -----------------------------------------------------
    const int m16  = lane & 15;        // A: M row / B: N col / store: col
    const int hi   = lane >> 4;        // 0 or 1
    const int koff = hi * 2;           // K sub-offset per ISA 16x4 fp32 layout
    const int mrow0 = row0 + wave * 16;
    const int lrbase = wave * 16 + hi * 8;

    const float* Asrc[3];
    Asrc[0] = xagg_in  + ((long)(mrow0 + m16)) * DIN + koff;
    Asrc[1] = xagg_out + ((long)(mrow0 + m16)) * DIN + koff;
    Asrc[2] = X        + ((long)(mrow0 + m16)) * DIN + koff;
    const float* Wsrc[3] = { w_in, w_out, w_loop };
    const float* Bias[2] = { b_in, b_out };

    #pragma unroll
    for (int s = 0; s < 3; ++s) {
        // Load this source's full A-tile into registers (90 x v2f = 180 VGPRs)
        v2f areg[KSTEPS];
        {
            const float* ap = Asrc[s];
            #pragma unroll
            for (int kk = 0; kk < KSTEPS; ++kk)
                areg[kk] = *(const v2f*)(ap + 4 * kk);
        }

        for (int ct = 0; ct < CT_COUNT; ++ct) {
            const int colbase = ct * 32;
            const int col0 = colbase + m16;
            const int col1 = col0 + 16;

            __syncthreads();
            // stage transposed weight chunk (branch-free clamped load;
            // consecutive threads read consecutive columns)
            for (int idx = tid; idx < 32 * DIN; idx += 256) {
                int nn = idx & 31;
                int kk = idx >> 5;
                int c  = colbase + nn;
                int cc = c < DOUT ? c : (DOUT - 1);
                float v = Wsrc[s][(long)kk * DOUT + cc];
                wlds[nn * DIN + kk] = (c < DOUT) ? v : 0.f;
            }
            __syncthreads();

            // prefetch next column tile's weights (global_prefetch_b8)
            if (ct + 1 < CT_COUNT) {
                for (int idx = tid * 16; idx < 32 * DIN; idx += 256 * 16) {
                    int nn = idx & 31;
                    int kk = idx >> 5;
                    int c  = colbase + 32 + nn;
                    if (c < DOUT)
                        __builtin_prefetch(&Wsrc[s][(long)kk * DOUT + c], 0, 1);
                }
            }

            v8f c0 = {}, c1 = {};
            const float* bp0 = &wlds[m16 * DIN + koff];
            const float* bp1 = bp0 + 16 * DIN;
            #pragma unroll
            for (int kk = 0; kk < KSTEPS; ++kk) {
                v2f b0 = *(const v2f*)(bp0 + 4 * kk);
                v2f b1 = *(const v2f*)(bp1 + 4 * kk);
                c0 = __builtin_amdgcn_wmma_f32_16x16x4_f32(
                        false, areg[kk], false, b0, (short)0, c0, false, false);
                c1 = __builtin_amdgcn_wmma_f32_16x16x4_f32(
                        false, areg[kk], false, b1, (short)0, c1, false, false);
            }

            // ---- epilogue: gate, accumulate into out (per-lane RMW) -----
            float sb0 = 0.f, sb1 = 0.f;
            if (s < 2) {
                sb0 = (col0 < DOUT) ? Bias[s][col0] : 0.f;
                sb1 = (col1 < DOUT) ? Bias[s][col1] : 0.f;
            }
            if (col0 < DOUT) {
                #pragma unroll
                for (int v = 0; v < 8; ++v) {
                    float sgv = s_sig[s][lrbase + v];
                    float t = c0[v];
                    if (s < 2) t += s_deg[s][lrbase + v] * sb0;
                    t *= sgv;
                    long row = (long)mrow0 + hi * 8 + v;
                    float* po = &out[row * DOUT + col0];
                    if (s == 0)      *po = t;
                    else if (s == 1) *po += t;
                    else { float r = *po + t; *po = r > 0.f ? r : 0.f; }
                }
            }
            if (col1 < DOUT) {
                #pragma unroll
                for (int v = 0; v < 8; ++v) {
                    float sgv = s_sig[s][lrbase + v];
                    float t = c1[v];
                    if (s < 2) t += s_deg[s][lrbase + v] * sb1;
                    t *= sgv;
                    long row = (long)mrow0 + hi * 8 + v;
                    float* po = &out[row * DOUT + col1];
                    if (s == 0)      *po = t;
                    else if (s == 1) *po += t;
                    else { float r = *po + t; *po = r > 0.f ? r : 0.f; }
                }
            }
        }
    }
}

// ---------------------------------------------------------------------------
extern "C" void kernel_launch(void* const* d_in, const int* in_sizes, int n_in,
                              void* d_out, int out_size, void* d_ws, size_t ws_size,
                              hipStream_t stream) {
    const float*     gcn_in  = (const float*)d_in[0];
    const long long* adj_ind = (const long long*)d_in[1];
    const float*     adjdata = (const float*)d_in[2];
    const float*     w_in    = (const float*)d_in[3];
    const float*     b_in    = (const float*)d_in[4];
    const float*     w_out   = (const float*)d_in[5];
    const float*     b_out   = (const float*)d_in[6];
    const float*     w_loop  = (const float*)d_in[7];
    const float*     w_gin   = (const float*)d_in[8];
    const float*     b_gin   = (const float*)d_in[9];
    const float*     w_gout  = (const float*)d_in[10];
    const float*     b_gout  = (const float*)d_in[11];
    const float*     w_gloop = (const float*)d_in[12];
    float* out = (float*)d_out;

    // workspace layout
    float* xagg_in  = (float*)d_ws;
    float* xagg_out = xagg_in  + (long)MROWS * DIN;
    float* deg_in   = xagg_out + (long)MROWS * DIN;
    float* deg_out  = deg_in   + MROWS;
    const long total_ws_floats = 2L * MROWS * DIN + 2L * MROWS;  // 36,966,400

    // 0) zero aggregation scratch (atomically accumulated below)
    zero_ws_kernel<<<4096, 256, 0, stream>>>((float4*)d_ws, total_ws_floats / 4);

    // 1) edge aggregation: one wave32 per edge
    {
        const long waves = (long)BB * EE;             // 102400
        const long threads = waves * 32;
        const int blocks = (int)((threads + 255) / 256);
        sp_agg_kernel<<<blocks, 256, 0, stream>>>(gcn_in, adj_ind, adjdata,
                                                  xagg_in, xagg_out, deg_in, deg_out);
    }

    // 2) fused triple GEMM + gates + epilogue
    {
        const int blocks = MROWS / ROWS_PER_BLOCK;    // 400
        fused_gemm_kernel<<<blocks, 256, 0, stream>>>(
            gcn_in, xagg_in, xagg_out, deg_in, deg_out,
            w_in, b_in, w_out, b_out, w_loop,
            w_gin, b_gin, w_gout, b_gout, w_gloop, out);
    }
}